// Transformer_47485158424639
// MI455X (gfx1250) — compile-verified
//
#include <hip/hip_runtime.h>
#include <math.h>

// ---------------------------------------------------------------------------
// TransGAN-style Swin generator forward, fp32, CDNA5 (gfx1250, wave32).
// All dense math runs on V_WMMA_F32_16X16X4_F32 (the fp32 tensor path).
// ---------------------------------------------------------------------------

typedef __attribute__((ext_vector_type(2))) float v2f;
typedef __attribute__((ext_vector_type(8))) float v8f;

#define GEMM_BIAS  1
#define GEMM_GELU  2
#define GEMM_ACCUM 4

static __device__ __forceinline__ v8f wmma4(v2f a, v2f b, v8f c) {
  // D = A(16x4) * B(4x16) + C(16x16), fp32
  return __builtin_amdgcn_wmma_f32_16x16x4_f32(false, a, false, b, (short)0, c,
                                               false, false);
}

// ---------------------------------------------------------------------------
// Tiled WMMA GEMM:  out[M,N] (=/+=) act(A[M,K] @ W[K,N] + bias)
// Block = 128 threads (4 waves) -> 128x128 C tile; each wave owns 64x64
// (4x4 fragments, 16 v8f accumulators). BK=32. Both LDS tiles are stored
// K-contiguous with stride 33 so every WMMA fragment is one b64 LDS read
// hitting 16 distinct banks. K must be a multiple of 32 (true here).
// ---------------------------------------------------------------------------
__global__ __launch_bounds__(128) void gemm_kernel(
    const float* __restrict__ A, const float* __restrict__ Wt,
    const float* __restrict__ bias, float* __restrict__ Cc,
    int M, int N, int K, int mode) {
  __shared__ float As[128 * 33];  // [m][k], stride 33
  __shared__ float Bt[128 * 33];  // [n][k] (transposed), stride 33

  const int tid = threadIdx.x;
  const int wave = tid >> 5;
  const int lane = tid & 31;
  const int lane16 = lane & 15;
  const int lh = lane >> 4;
  const int bm = blockIdx.y * 128;
  const int bn = blockIdx.x * 128;
  const int wr = wave >> 1;  // wave row (0..1)
  const int wc = wave & 1;   // wave col (0..1)

  v8f acc[4][4] = {};

  for (int kt = 0; kt < K; kt += 32) {
    // Speculative prefetch of the next K tile (global_prefetch_b8).
    if (kt + 32 < K) {
      int gr = bm + tid;
      if (gr >= M) gr = M - 1;
      __builtin_prefetch(&A[(size_t)gr * K + kt + 32], 0, 1);
      __builtin_prefetch(&Wt[(size_t)(kt + 32 + (tid & 31)) * N + bn +
                             ((tid >> 5) << 5)], 0, 1);
    }
    // Cooperative tile loads: 4096 floats each, 32 per thread, lane-major
    // so each wave touches 32 consecutive floats per step (coalesced).
#pragma unroll
    for (int u = 0; u < 32; ++u) {
      int i = tid + 128 * u;
      int r = i >> 5, c = i & 31;
      int gr = bm + r;
      As[r * 33 + c] = (gr < M) ? A[(size_t)gr * K + kt + c] : 0.0f;
    }
#pragma unroll
    for (int u = 0; u < 32; ++u) {
      int i = tid + 128 * u;
      int r = i >> 7, c = i & 127;  // r = k row, c = n col
      int gc = bn + c;
      Bt[c * 33 + r] = (gc < N) ? Wt[(size_t)(kt + r) * N + gc] : 0.0f;
    }
    __syncthreads();

#pragma unroll
    for (int kk = 0; kk < 32; kk += 4) {
      v2f a[4], b[4];
#pragma unroll
      for (int mi = 0; mi < 4; ++mi) {
        int m = wr * 64 + mi * 16 + lane16;
        const float* p = &As[m * 33 + kk + 2 * lh];
        a[mi][0] = p[0];
        a[mi][1] = p[1];
      }
#pragma unroll
      for (int ni = 0; ni < 4; ++ni) {
        int n = wc * 64 + ni * 16 + lane16;
        const float* p = &Bt[n * 33 + kk + 2 * lh];
        b[ni][0] = p[0];
        b[ni][1] = p[1];
      }
#pragma unroll
      for (int mi = 0; mi < 4; ++mi)
#pragma unroll
        for (int ni = 0; ni < 4; ++ni)
          acc[mi][ni] = wmma4(a[mi], b[ni], acc[mi][ni]);
    }
    __syncthreads();
  }

  // Epilogue. C/D layout: frag elem r -> row = r + 8*lh, col = lane16.
#pragma unroll
  for (int mi = 0; mi < 4; ++mi)
#pragma unroll
    for (int ni = 0; ni < 4; ++ni)
#pragma unroll
      for (int r = 0; r < 8; ++r) {
        int gm = bm + wr * 64 + mi * 16 + r + 8 * lh;
        int gn = bn + wc * 64 + ni * 16 + lane16;
        if (gm < M && gn < N) {
          float v = acc[mi][ni][r];
          if (mode & GEMM_BIAS) v += bias[gn];
          if (mode & GEMM_GELU)
            v = 0.5f * v * (1.0f + erff(v * 0.70710678118654752f));
          size_t o = (size_t)gm * N + gn;
          if (mode & GEMM_ACCUM) Cc[o] += v; else Cc[o] = v;
        }
      }
}

// ---------------------------------------------------------------------------
// Fused windowed attention for one (window, head). N=64 tokens, hd=128.
// S = scale*Q@K^T + rpb  -> softmax -> O = P@V, both matmuls on WMMA.
// LDS: one 8448-float buffer holds K as [token][dim] (stride 130) for the
// S phase, then is overwritten with V transposed [dim][token] (stride 66)
// so both phases use contiguous b64 fragment loads. Scores 64x65. ~50KB.
// ---------------------------------------------------------------------------
__global__ __launch_bounds__(128) void attn_kernel(
    const float* __restrict__ qkv,  // (nW*64, 3*C) ; per row [3][heads][128]
    const float* __restrict__ rpb,  // (225, heads)
    float* __restrict__ out,        // (nW*64, C)
    int C, int nheads) {
  __shared__ float KVb[128 * 66];  // 8448 floats, dual-layout (see above)
  __shared__ float Ss[64 * 65];

  const int h = blockIdx.x;
  const int wdw = blockIdx.y;
  const int tid = threadIdx.x;
  const int wave = tid >> 5;
  const int lane = tid & 31;
  const int lane16 = lane & 15;
  const int lh = lane >> 4;

  const size_t stride = 3 * (size_t)C;
  const float* Qg = qkv + (size_t)wdw * 64 * stride + (size_t)h * 128;
  const float* Kg = Qg + C;
  const float* Vg = Qg + 2 * C;

  // Stage K into LDS (row = key token, col = head dim), stride 130.
  for (int i = tid; i < 64 * 128; i += 128) {
    int r = i >> 7, cc = i & 127;
    KVb[r * 130 + cc] = Kg[(size_t)r * stride + cc];
  }
  __syncthreads();

  // ---- S = Q @ K^T : each wave computes a 32x32 block of the 64x64 scores.
  const int wr = wave >> 1, wc = wave & 1;
  v8f sc[2][2] = {};
  for (int kk = 0; kk < 128; kk += 4) {
    v2f a[2], b[2];
#pragma unroll
    for (int mi = 0; mi < 2; ++mi) {
      int m = wr * 32 + mi * 16 + lane16;  // query token
      const float* qp = Qg + (size_t)m * stride + kk + 2 * lh;
      a[mi][0] = qp[0];
      a[mi][1] = qp[1];
    }
#pragma unroll
    for (int ni = 0; ni < 2; ++ni) {
      int n = wc * 32 + ni * 16 + lane16;  // key token
      const float* p = &KVb[n * 130 + kk + 2 * lh];
      b[ni][0] = p[0];
      b[ni][1] = p[1];
    }
#pragma unroll
    for (int mi = 0; mi < 2; ++mi)
#pragma unroll
      for (int ni = 0; ni < 2; ++ni)
        sc[mi][ni] = wmma4(a[mi], b[ni], sc[mi][ni]);
  }
  const float scale = 0.08838834764831845f;  // 128^-0.5
#pragma unroll
  for (int mi = 0; mi < 2; ++mi)
#pragma unroll
    for (int ni = 0; ni < 2; ++ni)
#pragma unroll
      for (int r = 0; r < 8; ++r) {
        int row = wr * 32 + mi * 16 + r + 8 * lh;
        int col = wc * 32 + ni * 16 + lane16;
        Ss[row * 65 + col] = sc[mi][ni][r] * scale;
      }
  __syncthreads();

  // Stage V transposed into LDS: Vt[dim][token], stride 66 (K no longer
  // needed). Write banks: addr = 66*dim + tok -> 2*dim mod 64, distinct
  // across the 32 consecutive dims a wave writes per step.
  for (int i = tid; i < 64 * 128; i += 128) {
    int r = i >> 7, cc = i & 127;
    KVb[cc * 66 + r] = Vg[(size_t)r * stride + cc];
  }

  // Softmax with Swin relative-position bias (one thread per query row).
  if (tid < 64) {
    const int i = tid;
    const int yi = i >> 3, xi = i & 7;
    float mx = -1e30f;
    for (int j = 0; j < 64; ++j) {
      int yj = j >> 3, xj = j & 7;
      int ridx = (yi - yj + 7) * 15 + (xi - xj + 7);
      float v = Ss[i * 65 + j] + rpb[ridx * nheads + h];
      Ss[i * 65 + j] = v;
      mx = fmaxf(mx, v);
    }
    float sum = 0.0f;
    for (int j = 0; j < 64; ++j) {
      float e = expf(Ss[i * 65 + j] - mx);
      Ss[i * 65 + j] = e;
      sum += e;
    }
    float inv = 1.0f / sum;
    for (int j = 0; j < 64; ++j) Ss[i * 65 + j] *= inv;
  }
  __syncthreads();

  // ---- O = P @ V : wave w owns rows [16w, 16w+16), all 128 cols (8 frags).
  // A frag from Ss (contiguous), B frag from Vt (contiguous).
  v8f oc[8] = {};
  for (int kk = 0; kk < 64; kk += 4) {
    v2f a;
    int m = wave * 16 + lane16;
    const float* ap = &Ss[m * 65 + kk + 2 * lh];
    a[0] = ap[0];
    a[1] = ap[1];
#pragma unroll
    for (int ni = 0; ni < 8; ++ni) {
      v2f b;
      int n = ni * 16 + lane16;  // output dim
      const float* bp = &KVb[n * 66 + kk + 2 * lh];
      b[0] = bp[0];
      b[1] = bp[1];
      oc[ni] = wmma4(a, b, oc[ni]);
    }
  }
#pragma unroll
  for (int ni = 0; ni < 8; ++ni)
#pragma unroll
    for (int r = 0; r < 8; ++r) {
      int m = wave * 16 + r + 8 * lh;
      int n = ni * 16 + lane16;
      out[((size_t)wdw * 64 + m) * C + (size_t)h * 128 + n] = oc[ni][r];
    }
}

// ---------------------------------------------------------------------------
// LayerNorm: one wave (32 lanes) per token.
// ---------------------------------------------------------------------------
__global__ __launch_bounds__(256) void layernorm_kernel(
    const float* __restrict__ x, const float* __restrict__ g,
    const float* __restrict__ b, float* __restrict__ y, int T, int C) {
  const int wave = threadIdx.x >> 5;
  const int lane = threadIdx.x & 31;
  const int t = blockIdx.x * 8 + wave;
  if (t >= T) return;
  const float* row = x + (size_t)t * C;
  float s = 0.0f;
  for (int c = lane; c < C; c += 32) s += row[c];
#pragma unroll
  for (int off = 16; off > 0; off >>= 1) s += __shfl_xor(s, off, 32);
  const float mean = s / (float)C;
  float v = 0.0f;
  for (int c = lane; c < C; c += 32) { float d = row[c] - mean; v += d * d; }
#pragma unroll
  for (int off = 16; off > 0; off >>= 1) v += __shfl_xor(v, off, 32);
  const float inv = rsqrtf(v / (float)C + 1e-5f);
  float* yr = y + (size_t)t * C;
  for (int c = lane; c < C; c += 32)
    yr[c] = (row[c] - mean) * inv * g[c] + b[c];
}

// ---------------------------------------------------------------------------
// Window partition with fused pos-embed add, and window reverse.
// ---------------------------------------------------------------------------
__global__ void winpart_pos_kernel(const float* __restrict__ x,
                                   const float* __restrict__ pos,
                                   float* __restrict__ xw,
                                   int Bb, int H, int W, int C) {
  size_t total = (size_t)Bb * H * W * C;
  size_t i = (size_t)blockIdx.x * blockDim.x + threadIdx.x;
  if (i >= total) return;
  int c = (int)(i % C);
  size_t t = i / C;
  int b = (int)(t / ((size_t)H * W));
  int l = (int)(t % ((size_t)H * W));
  int yy = l / W, xx = l % W;
  int wy = yy >> 3, wx = xx >> 3, ty = yy & 7, tx = xx & 7;
  size_t widx = ((size_t)b * (H >> 3) + wy) * (size_t)(W >> 3) + wx;
  size_t orow = widx * 64 + (size_t)(ty * 8 + tx);
  xw[orow * C + c] = x[i] + pos[(size_t)l * C + c];
}

__global__ void winrev_kernel(const float* __restrict__ xw,
                              float* __restrict__ x,
                              int Bb, int H, int W, int C) {
  size_t total = (size_t)Bb * H * W * C;
  size_t i = (size_t)blockIdx.x * blockDim.x + threadIdx.x;
  if (i >= total) return;
  int c = (int)(i % C);
  size_t t = i / C;
  int b = (int)(t / ((size_t)H * W));
  int l = (int)(t % ((size_t)H * W));
  int yy = l / W, xx = l % W;
  int wy = yy >> 3, wx = xx >> 3, ty = yy & 7, tx = xx & 7;
  size_t widx = ((size_t)b * (H >> 3) + wy) * (size_t)(W >> 3) + wx;
  size_t orow = widx * 64 + (size_t)(ty * 8 + tx);
  x[i] = xw[orow * C + c];
}

// ---------------------------------------------------------------------------
// Bicubic x2 (Keys a=-0.5, half-pixel centers => fixed weights), token-major.
// ---------------------------------------------------------------------------
__global__ void bicubic_kernel(const float* __restrict__ x,
                               float* __restrict__ up,
                               int Bb, int H, int W, int C) {
  const int OH = 2 * H, OW = 2 * W;
  size_t total = (size_t)Bb * OH * OW * C;
  size_t idx = (size_t)blockIdx.x * blockDim.x + threadIdx.x;
  if (idx >= total) return;
  int c = (int)(idx % C);
  size_t t = idx / C;
  int b = (int)(t / ((size_t)OH * OW));
  int l = (int)(t % ((size_t)OH * OW));
  int oy = l / OW, ox = l % OW;
  const float W4[4] = {-0.0234375f, 0.2265625f, 0.8671875f, -0.0703125f};
  int py = oy & 1, px = ox & 1;
  int iy = oy >> 1, ix = ox >> 1;
  float acc = 0.0f;
#pragma unroll
  for (int j = 0; j < 4; ++j) {
    int sy = iy - 2 + py + j;
    sy = sy < 0 ? 0 : (sy > H - 1 ? H - 1 : sy);
    float wy = py ? W4[3 - j] : W4[j];
    float rowacc = 0.0f;
#pragma unroll
    for (int k = 0; k < 4; ++k) {
      int sx = ix - 2 + px + k;
      sx = sx < 0 ? 0 : (sx > W - 1 ? W - 1 : sx);
      float wx = px ? W4[3 - k] : W4[k];
      rowacc += wx * x[((size_t)b * H * W + (size_t)sy * W + sx) * C + c];
    }
    acc += wy * rowacc;
  }
  up[idx] = acc;
}

// ---------------------------------------------------------------------------
// Final 1x1 conv to RGB, channel-major output (B,3,64,64).
// ---------------------------------------------------------------------------
__global__ void rgb_kernel(const float* __restrict__ x,
                           const float* __restrict__ w,
                           const float* __restrict__ bias,
                           float* __restrict__ out) {
  int pix = blockIdx.x * blockDim.x + threadIdx.x;
  if (pix >= 16 * 4096) return;
  int b = pix >> 12, l = pix & 4095;
  const float* row = x + (size_t)pix * 256;
  float a0 = bias[0], a1 = bias[1], a2 = bias[2];
  for (int c = 0; c < 256; ++c) {
    float v = row[c];
    a0 += v * w[c * 3 + 0];
    a1 += v * w[c * 3 + 1];
    a2 += v * w[c * 3 + 2];
  }
  out[((size_t)b * 3 + 0) * 4096 + l] = a0;
  out[((size_t)b * 3 + 1) * 4096 + l] = a1;
  out[((size_t)b * 3 + 2) * 4096 + l] = a2;
}

// ---------------------------------------------------------------------------
// Host orchestration.
// Input flattening assumed = recursive dict-insertion order of setup_inputs():
//   0:z 1:l1_w 2:l1_b 3..6:pos_embed[0..3]
//   7 + (s*2+d)*12 + {0:n1_g 1:n1_b 2:qkv_w 3:proj_w 4:proj_b 5:rpb
//                     6:n2_g 7:n2_b 8:fc1_w 9:fc1_b 10:fc2_w 11:fc2_b}
//   103..108: up[0].w,b up[1].w,b up[2].w,b    109:rgb_w 110:rgb_b
// ---------------------------------------------------------------------------
extern "C" void kernel_launch(void* const* d_in, const int* in_sizes, int n_in,
                              void* d_out, int out_size, void* d_ws,
                              size_t ws_size, hipStream_t stream) {
  (void)in_sizes; (void)n_in; (void)out_size; (void)ws_size;
  const float* z    = (const float*)d_in[0];
  const float* l1w  = (const float*)d_in[1];
  const float* l1b  = (const float*)d_in[2];
  const float* pos[4] = {(const float*)d_in[3], (const float*)d_in[4],
                         (const float*)d_in[5], (const float*)d_in[6]};
  const float* upw[3] = {(const float*)d_in[103], (const float*)d_in[105],
                         (const float*)d_in[107]};
  const float* upb[3] = {(const float*)d_in[104], (const float*)d_in[106],
                         (const float*)d_in[108]};
  const float* rgbw = (const float*)d_in[109];
  const float* rgbb = (const float*)d_in[110];

  // Workspace layout (floats): A/B activation ping-pong, LN/attn buf, big tmp.
  float* AB = (float*)d_ws;                   // 16.78M floats
  float* BB = AB + 16777216ull;               // 16.78M floats
  float* LN = BB + 16777216ull;               // 16.78M floats
  float* TM = LN + 16777216ull;               // 67.11M floats

  static const int DIMS[4]  = {2048, 1024, 512, 256};
  static const int HEADS[4] = {16, 8, 4, 2};
  const dim3 blk(128);

  // z @ l1_w + l1_b : (16,256)x(256,131072) -> tokens (1024, 2048) in AB.
  gemm_kernel<<<dim3(131072 / 128, 1), blk, 0, stream>>>(
      z, l1w, l1b, AB, 16, 131072, 256, GEMM_BIAS);

  int H = 8;
  for (int s = 0; s < 4; ++s) {
    const int D = DIMS[s];
    const int hds = HEADS[s];
    const int T = 16 * H * H;     // total tokens
    const int nW = T / 64;        // total windows
    const size_t tot = (size_t)T * D;

    winpart_pos_kernel<<<(unsigned)((tot + 255) / 256), 256, 0, stream>>>(
        AB, pos[s], BB, 16, H, H, D);

    for (int d = 0; d < 2; ++d) {
      const int bi = 7 + (s * 2 + d) * 12;
      const float* n1g = (const float*)d_in[bi + 0];
      const float* n1b = (const float*)d_in[bi + 1];
      const float* qkvw = (const float*)d_in[bi + 2];
      const float* projw = (const float*)d_in[bi + 3];
      const float* projb = (const float*)d_in[bi + 4];
      const float* rpb = (const float*)d_in[bi + 5];
      const float* n2g = (const float*)d_in[bi + 6];
      const float* n2b = (const float*)d_in[bi + 7];
      const float* fc1w = (const float*)d_in[bi + 8];
      const float* fc1b = (const float*)d_in[bi + 9];
      const float* fc2w = (const float*)d_in[bi + 10];
      const float* fc2b = (const float*)d_in[bi + 11];

      layernorm_kernel<<<(T + 7) / 8, 256, 0, stream>>>(BB, n1g, n1b, LN, T, D);
      gemm_kernel<<<dim3((3 * D + 127) / 128, (T + 127) / 128), blk, 0,
                    stream>>>(LN, qkvw, nullptr, TM, T, 3 * D, D, 0);
      attn_kernel<<<dim3(hds, nW), blk, 0, stream>>>(TM, rpb, LN, D, hds);
      gemm_kernel<<<dim3((D + 127) / 128, (T + 127) / 128), blk, 0, stream>>>(
          LN, projw, projb, BB, T, D, D, GEMM_BIAS | GEMM_ACCUM);

      layernorm_kernel<<<(T + 7) / 8, 256, 0, stream>>>(BB, n2g, n2b, LN, T, D);
      gemm_kernel<<<dim3((4 * D + 127) / 128, (T + 127) / 128), blk, 0,
                    stream>>>(LN, fc1w, fc1b, TM, T, 4 * D, D,
                              GEMM_BIAS | GEMM_GELU);
      gemm_kernel<<<dim3((D + 127) / 128, (T + 127) / 128), blk, 0, stream>>>(
          TM, fc2w, fc2b, BB, T, D, 4 * D, GEMM_BIAS | GEMM_ACCUM);
    }

    winrev_kernel<<<(unsigned)((tot + 255) / 256), 256, 0, stream>>>(
        BB, AB, 16, H, H, D);

    if (s < 3) {
      const size_t utot = (size_t)16 * 4 * H * H * D;
      bicubic_kernel<<<(unsigned)((utot + 255) / 256), 256, 0, stream>>>(
          AB, TM, 16, H, H, D);
      const int M2 = 16 * 4 * H * H;
      gemm_kernel<<<dim3((D / 2 + 127) / 128, (M2 + 127) / 128), blk, 0,
                    stream>>>(TM, upw[s], upb[s], AB, M2, D / 2, D, GEMM_BIAS);
      H *= 2;
    }
  }

  rgb_kernel<<<(16 * 4096 + 255) / 256, 256, 0, stream>>>(
      AB, rgbw, rgbb, (float*)d_out);
}